// GenGNN_16887811408662
// MI455X (gfx1250) — compile-verified
//
#include <hip/hip_runtime.h>
#include <math.h>

// ---------------------------------------------------------------------------
// GNN link predictor for MI455X (gfx1250, wave32).
//  - Dense GEMMs: V_WMMA_F32_16X16X4_F32 (exact fp32 on the matrix pipe).
//    Each wave computes a 16x(16*TN) strip so the A fragment is reused TN
//    times per K-step (workload is L2-resident / latency-bound, so fewer
//    VMEM issues per WMMA is the win; no precision downcast needed).
//  - GCN aggregation: native global_atomic_add_f32 scatter, coalesced per
//    wave (consecutive lanes = consecutive features of one edge).
//  - Edge decoder: wave-per-edge cooperative dot product + shfl_xor butterfly.
// ---------------------------------------------------------------------------

typedef __attribute__((ext_vector_type(2))) float v2f;
typedef __attribute__((ext_vector_type(8))) float v8f;

// ---------------------------------------------------------------------------
// fp32 WMMA GEMM: C[M,N] = A[M,K] @ B[K,N] (+bias, optional ReLU).
// One wave computes a 16 x (16*TN) strip, stepping K by 4.
// Requires: M%16==0, N%(16*TN)==0, K%4==0. Here M=50000=3125*16,
// N in {128,64} -> TN=4, N=16 -> TN=1, K in {256,128}.
//
// Lane layouts per CDNA5 ISA 7.12.2 (wave32):
//   A 16x4 f32 : lane l<16 -> M=l, VGPRs{0,1}=K{0,1}; lanes 16..31 -> K{2,3}
//   B 4x16 f32 : lane l<16 -> N=l, VGPRs{0,1}=K{0,1}; lanes 16..31 -> K{2,3}
//   C/D 16x16  : VGPR v, lane l<16 -> (M=v, N=l); lane>=16 -> (M=v+8, N=l-16)
// ---------------------------------------------------------------------------
template <int TN>
__global__ void k_wmma_gemm(const float* __restrict__ A,
                            const float* __restrict__ B,
                            const float* __restrict__ bias,
                            float* __restrict__ C,
                            int M, int N, int K, int relu)
{
    const int wave = blockIdx.x * (blockDim.x >> 5) + (threadIdx.x >> 5);
    const int lane = threadIdx.x & 31;
    const int strips_n = N / (16 * TN);
    const int tiles_m  = M >> 4;
    if (wave >= tiles_m * strips_n) return;   // wave-uniform: EXEC stays all-1s

    const int tm = wave / strips_n;
    const int ts = wave - tm * strips_n;
    const int half = lane >> 4;               // K pair {0,1} vs {2,3}
    const int l    = lane & 15;

    const float* __restrict__ Arow  = A + (size_t)(tm * 16 + l) * K;
    const float* __restrict__ Bbase = B + ts * 16 * TN + l;

    v8f acc[TN];
#pragma unroll
    for (int t = 0; t < TN; ++t) {
        v8f z = {};
        acc[t] = z;
    }

    for (int k0 = 0; k0 < K; k0 += 4) {
        const int ka = k0 + half * 2;
        v2f a;
        a.x = Arow[ka];                       // contiguous pair -> b64 load
        a.y = Arow[ka + 1];
        const float* __restrict__ Brow0 = Bbase + (size_t)ka * N;
        const float* __restrict__ Brow1 = Bbase + (size_t)(ka + 1) * N;
#pragma unroll
        for (int t = 0; t < TN; ++t) {        // A fragment reused TN times
            v2f b;
            b.x = Brow0[t * 16];
            b.y = Brow1[t * 16];
            acc[t] = __builtin_amdgcn_wmma_f32_16x16x4_f32(
                /*neg_a=*/false, a, /*neg_b=*/false, b,
                /*c_mod=*/(short)0, acc[t], /*reuse_a=*/false, /*reuse_b=*/false);
        }
    }

#pragma unroll
    for (int t = 0; t < TN; ++t) {
        const int col = ts * 16 * TN + t * 16 + l;
        const float bv = bias ? bias[col] : 0.0f;
#pragma unroll
        for (int v = 0; v < 8; ++v) {
            const int row = tm * 16 + v + half * 8;
            float x = acc[t][v] + bv;
            if (relu) x = fmaxf(x, 0.0f);
            C[(size_t)row * N + col] = x;
        }
    }
}

// ---------------------------------------------------------------------------
// GCN normalization: deg[i] = 1 (self loop) + #(dst==i); dinv = rsqrt(deg).
// ---------------------------------------------------------------------------
__global__ void k_init_deg(float* __restrict__ deg, int n)
{
    int i = blockIdx.x * blockDim.x + threadIdx.x;
    if (i < n) deg[i] = 1.0f;                // self loop
}

__global__ void k_count_deg(const int* __restrict__ dst, float* __restrict__ deg, int e)
{
    int i = blockIdx.x * blockDim.x + threadIdx.x;
    if (i < e) atomicAdd(&deg[dst[i]], 1.0f);
}

__global__ void k_deg_to_dinv(float* __restrict__ deg, int n)
{
    int i = blockIdx.x * blockDim.x + threadIdx.x;
    if (i < n) deg[i] = rsqrtf(deg[i]);      // deg >= 1 always (self loops)
}

// Self-loop term initializes aggregation buffer: agg[i] = xw[i] * dinv[i]^2
// (f = 1<<fshift features per node)
__global__ void k_selfloop(const float* __restrict__ xw, const float* __restrict__ dinv,
                           float* __restrict__ agg, int n, int fshift)
{
    long long t = (long long)blockIdx.x * blockDim.x + threadIdx.x;
    if (t >= ((long long)n << fshift)) return;
    int i = (int)(t >> fshift);
    float di = dinv[i];
    agg[t] = xw[t] * di * di;
}

// Edge scatter: agg[dst] += xw[src] * dinv[src]*dinv[dst]
// One thread per (edge, feature); lanes of a wave cover consecutive features
// of one edge -> coalesced gather + conflict-free f32 atomics within a wave.
__global__ void k_scatter(const int* __restrict__ src, const int* __restrict__ dst,
                          const float* __restrict__ xw, const float* __restrict__ dinv,
                          float* __restrict__ agg, int e, int fshift)
{
    long long t = (long long)blockIdx.x * blockDim.x + threadIdx.x;
    if (t >= ((long long)e << fshift)) return;
    int ed = (int)(t >> fshift);
    int c  = (int)(t & ((1 << fshift) - 1));
    int s = src[ed];
    int d = dst[ed];
    float v = xw[((size_t)s << fshift) + c] * dinv[s] * dinv[d];
    atomicAdd(&agg[((size_t)d << fshift) + c], v);
}

__global__ void k_bias_act(float* __restrict__ a, const float* __restrict__ bias,
                           int n, int fshift, int relu)
{
    long long t = (long long)blockIdx.x * blockDim.x + threadIdx.x;
    if (t >= ((long long)n << fshift)) return;
    int c = (int)(t & ((1 << fshift) - 1));
    float x = a[t] + bias[c];
    if (relu) x = fmaxf(x, 0.0f);
    a[t] = x;
}

// ---------------------------------------------------------------------------
// Per-node log_softmax over C classes + y_prob (one-hot for train nodes,
// exp(log_prob) otherwise). log-probs go straight into the output buffer.
// ---------------------------------------------------------------------------
__global__ void k_softmax_prob(const float* __restrict__ logits,
                               const int* __restrict__ y,
                               const unsigned char* __restrict__ mask, // jnp bool: 1 byte
                               float* __restrict__ lp_out,
                               float* __restrict__ yprob,
                               int n, int c)
{
    int i = blockIdx.x * blockDim.x + threadIdx.x;
    if (i >= n) return;
    const float* lg = logits + (size_t)i * c;
    float m = lg[0];
    for (int j = 1; j < c; ++j) m = fmaxf(m, lg[j]);
    float s = 0.0f;
    for (int j = 0; j < c; ++j) s += expf(lg[j] - m);
    float lse = m + logf(s);
    bool train = mask[i] != 0;
    int yi = y[i];
    for (int j = 0; j < c; ++j) {
        float lp = lg[j] - lse;
        lp_out[(size_t)i * c + j] = lp;
        yprob[(size_t)i * c + j] = train ? (j == yi ? 1.0f : 0.0f) : expf(lp);
    }
}

// ---------------------------------------------------------------------------
// Edge decoder, wave-per-edge: out[e] = [xe[s], xe[d], yp[s], yp[d]] . Wd + bd
// 32 lanes stream the 2*hx+2*c concat feature coalesced, then butterfly-reduce.
// ---------------------------------------------------------------------------
__global__ void k_decode_wave(const int* __restrict__ ei,  // [2,E]: row0=src, row1=dst
                              const float* __restrict__ xe,
                              const float* __restrict__ yp,
                              const float* __restrict__ Wd, // [2*(hx+c)]
                              const float* __restrict__ bd,
                              float* __restrict__ out, int e, int hx, int c)
{
    int wv   = blockIdx.x * (blockDim.x >> 5) + (threadIdx.x >> 5);
    int lane = threadIdx.x & 31;
    if (wv >= e) return;                      // wave-uniform
    int s = ei[wv];
    int d = ei[e + wv];
    const float* xs = xe + (size_t)s * hx;
    const float* xd = xe + (size_t)d * hx;
    const float* ys = yp + (size_t)s * c;
    const float* yd = yp + (size_t)d * c;
    const int total = 2 * hx + 2 * c;         // 160
    float acc = 0.0f;
    for (int idx = lane; idx < total; idx += 32) {
        float v;
        if (idx < hx)               v = xs[idx];
        else if (idx < 2 * hx)      v = xd[idx - hx];
        else if (idx < 2 * hx + c)  v = ys[idx - 2 * hx];
        else                        v = yd[idx - 2 * hx - c];
        acc += v * Wd[idx];
    }
#pragma unroll
    for (int off = 16; off > 0; off >>= 1)
        acc += __shfl_xor(acc, off, 32);
    if (lane == 0) out[wv] = acc + bd[0];
}

// ---------------------------------------------------------------------------
// Launch
// ---------------------------------------------------------------------------
static inline int cdiv(long long a, long long b) { return (int)((a + b - 1) / b); }

static inline int ilog2(int v) { int s = 0; while ((1 << s) < v) ++s; return s; }

static void launch_gemm(const float* A, const float* B, const float* bias, float* C,
                        int M, int N, int K, int relu, hipStream_t stream)
{
    if (N % 64 == 0) {
        int waves = (M / 16) * (N / 64);
        k_wmma_gemm<4><<<cdiv(waves, 4), 128, 0, stream>>>(A, B, bias, C, M, N, K, relu);
    } else {
        int waves = (M / 16) * (N / 16);
        k_wmma_gemm<1><<<cdiv(waves, 4), 128, 0, stream>>>(A, B, bias, C, M, N, K, relu);
    }
}

extern "C" void kernel_launch(void* const* d_in, const int* in_sizes, int n_in,
                              void* d_out, int out_size, void* d_ws, size_t ws_size,
                              hipStream_t stream)
{
    // Inputs (setup_inputs order)
    const float* x    = (const float*)d_in[0];
    const int*   ei   = (const int*)d_in[1];   // [2,E] pos edges
    const int*   nei  = (const int*)d_in[2];   // [2,E] neg edges
    const int*   y    = (const int*)d_in[3];
    const unsigned char* mask = (const unsigned char*)d_in[4]; // jnp bool -> 1 byte
    const float* W1   = (const float*)d_in[5];
    const float* b1   = (const float*)d_in[6];
    const float* W2   = (const float*)d_in[7];
    const float* b2   = (const float*)d_in[8];
    const float* Wx   = (const float*)d_in[9];
    const float* bx   = (const float*)d_in[10];
    const float* Wd   = (const float*)d_in[11];
    const float* bd   = (const float*)d_in[12];

    // Dimensions derived from in_sizes
    const int N  = in_sizes[3];          // 50000
    const int F  = in_sizes[0] / N;      // 256
    const int H  = in_sizes[6];          // 128
    const int C  = in_sizes[8];          // 16
    const int HX = in_sizes[10];         // 64
    const int E  = in_sizes[1] / 2;      // 500000

    const int hs = ilog2(H);             // 7
    const int cs = ilog2(C);             // 4

    const int* src = ei;
    const int* dst = ei + E;

    // Workspace layout (floats): ~74 MB total, L2-resident on MI455X (192 MB)
    float* w = (float*)d_ws;
    float* dinv   = w;                        // N      (deg -> dinv in place)
    float* xw1    = dinv + N;                 // N*H
    float* h      = xw1 + (size_t)N * H;      // N*H    (agg1 -> relu'd hidden)
    float* xw2    = h   + (size_t)N * H;      // N*C
    float* logits = xw2 + (size_t)N * C;      // N*C    (agg2 -> +bias)
    float* yprob  = logits + (size_t)N * C;   // N*C
    float* xe     = yprob  + (size_t)N * C;   // N*HX

    float* out_pos = (float*)d_out;           // E
    float* out_neg = out_pos + E;             // E
    float* out_lp  = out_neg + E;             // N*C

    const int BT = 256;

    // 1) degrees / normalization
    k_init_deg   <<<cdiv(N, BT), BT, 0, stream>>>(dinv, N);
    k_count_deg  <<<cdiv(E, BT), BT, 0, stream>>>(dst, dinv, E);
    k_deg_to_dinv<<<cdiv(N, BT), BT, 0, stream>>>(dinv, N);

    // 2) layer 1: xw1 = x @ W1 ; h = relu(aggregate(xw1) + b1)
    launch_gemm(x, W1, nullptr, xw1, N, H, F, 0, stream);
    k_selfloop<<<cdiv((long long)N << hs, BT), BT, 0, stream>>>(xw1, dinv, h, N, hs);
    k_scatter <<<cdiv((long long)E << hs, BT), BT, 0, stream>>>(src, dst, xw1, dinv, h, E, hs);
    k_bias_act<<<cdiv((long long)N << hs, BT), BT, 0, stream>>>(h, b1, N, hs, 1);

    // 3) layer 2: xw2 = h @ W2 ; logits = aggregate(xw2) + b2   (TEMP == 1)
    launch_gemm(h, W2, nullptr, xw2, N, C, H, 0, stream);
    k_selfloop<<<cdiv((long long)N << cs, BT), BT, 0, stream>>>(xw2, dinv, logits, N, cs);
    k_scatter <<<cdiv((long long)E << cs, BT), BT, 0, stream>>>(src, dst, xw2, dinv, logits, E, cs);
    k_bias_act<<<cdiv((long long)N << cs, BT), BT, 0, stream>>>(logits, b2, N, cs, 0);

    // 4) log_softmax + y_prob
    k_softmax_prob<<<cdiv(N, BT), BT, 0, stream>>>(logits, y, mask, out_lp, yprob, N, C);

    // 5) node feature encoder: xe = relu(x @ Wx + bx)
    launch_gemm(x, Wx, bx, xe, N, HX, F, 1, stream);

    // 6) edge decoders (wave per edge, 8 waves per block)
    k_decode_wave<<<cdiv((long long)E * 32, BT), BT, 0, stream>>>(ei,  xe, yprob, Wd, bd, out_pos, E, HX, C);
    k_decode_wave<<<cdiv((long long)E * 32, BT), BT, 0, stream>>>(nei, xe, yprob, Wd, bd, out_neg, E, HX, C);
}